// GliomaGNN_22497038696708
// MI455X (gfx1250) — compile-verified
//
#include <hip/hip_runtime.h>
#include <cstdint>
#include <cstddef>

// ---------------------------------------------------------------------------
// GliomaGNN forward on gfx1250 (MI455X): 2-layer hetero GAT + MLP heads.
// Dense GEMMs use V_WMMA_F32_16X16X4_F32 (fp32 matrix path, exact-precision)
// with weight (B) fragments held register-resident per wave.
// Edge softmax/aggregation uses 3-pass segment ops with L2-resident atomics.
// ---------------------------------------------------------------------------

typedef __attribute__((ext_vector_type(2))) float v2f;
typedef __attribute__((ext_vector_type(8))) float v8f;

// float atomic max via monotonic int/uint punning (init value must be -inf)
__device__ __forceinline__ void atomicMaxF(float* addr, float val) {
  if (val >= 0.0f) atomicMax((int*)addr, __float_as_int(val));
  else             atomicMin((unsigned int*)addr, __float_as_uint(val));
}

__global__ void fill_kernel(float* __restrict__ p, float v, size_t n) {
  size_t i = (size_t)blockIdx.x * blockDim.x + threadIdx.x;
  size_t stride = (size_t)gridDim.x * blockDim.x;
  for (; i < n; i += stride) p[i] = v;
}

// ---------------------------------------------------------------------------
// GEMM: Out[M,N] = A[M,K] @ W[K,N] (+bias) (+relu), via V_WMMA_F32_16X16X4_F32
// Each wave owns one 16-col strip (tileN): preloads all K/4 B fragments into
// registers once, then loops over MT consecutive 16-row tiles doing only
// A loads + WMMA + stores. 8 waves per 256-thread block (wave32).
// Requires M % 16 == 0 (NP=100000, NT=50000 both divisible by 16).
// A-frag (16x4 f32): lane<16 holds row=lane,K={0,1}; lane>=16 row=lane-16,K={2,3}
// B-frag (4x16 f32): mirrored; C/D: row = 8*(lane>=16)+i, col = lane&15.
// ---------------------------------------------------------------------------
template<int K, int N, bool BIAS, bool RELU, int MT>
__global__ __launch_bounds__(256) void gemm_wmma_kernel(
    const float* __restrict__ A, const float* __restrict__ W,
    const float* __restrict__ bias, float* __restrict__ Out, int M)
{
  constexpr int KS      = K / 4;    // k-steps (WMMA depth 4)
  constexpr int TILES_N = N / 16;
  const int lane   = threadIdx.x & 31;
  const int wave   = threadIdx.x >> 5;
  const int gw     = blockIdx.x * (blockDim.x >> 5) + wave;  // global wave id
  const int mGroup = gw / TILES_N;
  const int tileN  = gw % TILES_N;
  const int mtile0 = mGroup * MT;
  if (mtile0 * 16 >= M) return;              // wave-uniform exit: EXEC stays full

  const int mn   = lane & 15;                // row (for A) / col (for B,D)
  const int koff = (lane >> 4) * 2;          // lane-half K offset

  // --- preload all B fragments for this column strip (register-resident) ---
  const float* Wcol = W + tileN * 16 + mn;   // W row-major [K,N]
  v2f b[KS];
  #pragma unroll
  for (int s = 0; s < KS; ++s) {
    b[s].x = Wcol[(size_t)(4 * s + koff) * N];
    b[s].y = Wcol[(size_t)(4 * s + koff + 1) * N];
  }
  float bv = 0.0f;
  if (BIAS) bv = bias[tileN * 16 + mn];
  const int rbase = (lane >> 4) * 8;

  // --- loop over MT row tiles, A loads + WMMA chain + store ---
  #pragma unroll
  for (int t = 0; t < MT; ++t) {
    const int tm = mtile0 + t;
    if (tm * 16 >= M) break;                 // wave-uniform
    const float* Arow = A + (size_t)(tm * 16 + mn) * K;
    v8f acc = {};
    #pragma unroll
    for (int s = 0; s < KS; ++s) {
      v2f a;
      a.x = Arow[4 * s + koff];
      a.y = Arow[4 * s + koff + 1];
      acc = __builtin_amdgcn_wmma_f32_16x16x4_f32(
          false, a, false, b[s], (short)0, acc, false, false);
    }
    float* Orow = Out + (size_t)(tm * 16 + rbase) * N + tileN * 16 + mn;
    #pragma unroll
    for (int i = 0; i < 8; ++i) {
      float v = acc[i] + bv;
      if (RELU) v = fmaxf(v, 0.0f);
      Orow[(size_t)i * N] = v;
    }
  }
}

// a[node*H + h] = sum_c h[node, h*32+c] * av[h*32+c]
template<int H>
__global__ void attn_logits_kernel(const float* __restrict__ hfeat,
                                   const float* __restrict__ av,
                                   float* __restrict__ out, int n)
{
  int i = blockIdx.x * blockDim.x + threadIdx.x;
  if (i >= n * H) return;
  const float* hp = hfeat + (size_t)i * 32;
  const float* ap = av + (i % H) * 32;
  float s = 0.0f;
  #pragma unroll
  for (int c = 0; c < 32; ++c) s = fmaf(hp[c], ap[c], s);
  out[i] = s;
}

// pass A: segment max of leaky_relu(a_s[src]+a_d[dst]) per (dst, head)
template<int H>
__global__ void edge_max_kernel(const int* __restrict__ src, const int* __restrict__ dst,
                                const float* __restrict__ as_, const float* __restrict__ ad_,
                                float* __restrict__ mx, int E)
{
  int i = blockIdx.x * blockDim.x + threadIdx.x;
  if (i >= E) return;
  int s = src[i], d = dst[i];
  #pragma unroll
  for (int h = 0; h < H; ++h) {
    float e = as_[(size_t)s * H + h] + ad_[(size_t)d * H + h];
    e = e > 0.0f ? e : 0.2f * e;
    atomicMaxF(&mx[(size_t)d * H + h], e);
  }
}

// pass B: segment sum of exp(e - max)
template<int H>
__global__ void edge_expsum_kernel(const int* __restrict__ src, const int* __restrict__ dst,
                                   const float* __restrict__ as_, const float* __restrict__ ad_,
                                   const float* __restrict__ mx, float* __restrict__ den, int E)
{
  int i = blockIdx.x * blockDim.x + threadIdx.x;
  if (i >= E) return;
  int s = src[i], d = dst[i];
  #pragma unroll
  for (int h = 0; h < H; ++h) {
    float e = as_[(size_t)s * H + h] + ad_[(size_t)d * H + h];
    e = e > 0.0f ? e : 0.2f * e;
    atomicAdd(&den[(size_t)d * H + h], expf(e - mx[(size_t)d * H + h]));
  }
}

// pass C: out[dst,h,:] += attn * hs[src,h,:]   (one thread per edge*head)
template<int H>
__global__ void edge_scatter_kernel(const int* __restrict__ src, const int* __restrict__ dst,
                                    const float* __restrict__ as_, const float* __restrict__ ad_,
                                    const float* __restrict__ mx, const float* __restrict__ den,
                                    const float* __restrict__ hs, float* __restrict__ out, int E)
{
  int i = blockIdx.x * blockDim.x + threadIdx.x;
  int e = i / H, h = i % H;
  if (e >= E) return;
  int s = src[e], d = dst[e];
  float el = as_[(size_t)s * H + h] + ad_[(size_t)d * H + h];
  el = el > 0.0f ? el : 0.2f * el;
  float attn = expf(el - mx[(size_t)d * H + h]) / (den[(size_t)d * H + h] + 1e-16f);
  const float* hrow = hs + ((size_t)s * H + h) * 32;
  float* orow = out + ((size_t)d * H + h) * 32;
  #pragma unroll
  for (int c = 0; c < 32; ++c) atomicAdd(&orow[c], hrow[c] * attn);
}

template<bool RELU, int F>
__global__ void bias_act_kernel(float* __restrict__ x, const float* __restrict__ b, size_t total)
{
  size_t i = (size_t)blockIdx.x * blockDim.x + threadIdx.x;
  if (i >= total) return;
  float v = x[i] + b[i % F];
  if (RELU) v = fmaxf(v, 0.0f);
  x[i] = v;
}

// heads: 32 -> relu(16) -> 1, then sigmoid / relu / softplus
__global__ void heads_kernel(const float* __restrict__ h2t,
    const float* __restrict__ rW1, const float* __restrict__ rW2,
    const float* __restrict__ rb1, const float* __restrict__ rb2,
    const float* __restrict__ sW1, const float* __restrict__ sW2,
    const float* __restrict__ sb1, const float* __restrict__ sb2,
    const float* __restrict__ uW1, const float* __restrict__ uW2,
    const float* __restrict__ ub1, const float* __restrict__ ub2,
    float* __restrict__ resp, float* __restrict__ surv, float* __restrict__ unc, int NT)
{
  int t = blockIdx.x * blockDim.x + threadIdx.x;
  if (t >= NT) return;
  float h[32];
  #pragma unroll
  for (int c = 0; c < 32; ++c) h[c] = h2t[(size_t)t * 32 + c];
  auto mlp = [&](const float* W1, const float* b1, const float* W2, const float* b2) -> float {
    float y = b2[0];
    #pragma unroll
    for (int j = 0; j < 16; ++j) {
      float z = b1[j];
      #pragma unroll
      for (int c = 0; c < 32; ++c) z = fmaf(h[c], W1[c * 16 + j], z);
      y = fmaf(fmaxf(z, 0.0f), W2[j], y);
    }
    return y;
  };
  float r = mlp(rW1, rb1, rW2, rb2);
  resp[t] = 1.0f / (1.0f + expf(-r));
  float s = mlp(sW1, sb1, sW2, sb2);
  surv[t] = fmaxf(s, 0.0f);
  float u = mlp(uW1, ub1, uW2, ub2);
  unc[t] = fmaxf(u, 0.0f) + log1pf(expf(-fabsf(u)));   // stable softplus
}

// ---------------------------------------------------------------------------
// Launch orchestration
// d_in order (jax tree-flatten sorts dict keys, ASCII):
//  0 x_patient[NP,128]  1 x_treatment[NT,64]  2 ei_pt[2,E]  3 ei_tp[2,E]
//  4..8   c1_pt {Wd[32,128], Ws[32,128], ad[4,32], as[4,32], b[128]}
//  9..13  c1_tp {Wd, Ws, ad, as, b}
// 14..18  c2_pt {Wd[128,32], Ws[128,32], ad[32], as[32], b[32]}
// 19..23  c2_tp {Wd, Ws, ad, as, b}
// 24 enc_p_W[128,32] 25 enc_p_b[32] 26 enc_t_W[64,32] 27 enc_t_b[32]
// 28..31 resp {W1[32,16], W2[16], b1[16], b2[1]}   32..35 surv   36..39 unc
// d_out: resp[NT] | surv[NT] | unc[NT] | h2_p[NP*32] | h2_t[NT*32]
// ---------------------------------------------------------------------------
extern "C" void kernel_launch(void* const* d_in, const int* in_sizes, int n_in,
                              void* d_out, int out_size, void* d_ws, size_t ws_size,
                              hipStream_t stream) {
  const int NP = in_sizes[0] / 128;
  const int NT = in_sizes[1] / 64;
  const int E  = in_sizes[2] / 2;

  const float* x_p = (const float*)d_in[0];
  const float* x_t = (const float*)d_in[1];
  const int* ei_pt = (const int*)d_in[2];
  const int* ei_tp = (const int*)d_in[3];
  const float *c1pt_Wd = (const float*)d_in[4],  *c1pt_Ws = (const float*)d_in[5],
              *c1pt_ad = (const float*)d_in[6],  *c1pt_as = (const float*)d_in[7],
              *c1pt_b  = (const float*)d_in[8];
  const float *c1tp_Wd = (const float*)d_in[9],  *c1tp_Ws = (const float*)d_in[10],
              *c1tp_ad = (const float*)d_in[11], *c1tp_as = (const float*)d_in[12],
              *c1tp_b  = (const float*)d_in[13];
  const float *c2pt_Wd = (const float*)d_in[14], *c2pt_Ws = (const float*)d_in[15],
              *c2pt_ad = (const float*)d_in[16], *c2pt_as = (const float*)d_in[17],
              *c2pt_b  = (const float*)d_in[18];
  const float *c2tp_Wd = (const float*)d_in[19], *c2tp_Ws = (const float*)d_in[20],
              *c2tp_ad = (const float*)d_in[21], *c2tp_as = (const float*)d_in[22],
              *c2tp_b  = (const float*)d_in[23];
  const float *enc_p_W = (const float*)d_in[24], *enc_p_b = (const float*)d_in[25];
  const float *enc_t_W = (const float*)d_in[26], *enc_t_b = (const float*)d_in[27];
  const float *rW1 = (const float*)d_in[28], *rW2 = (const float*)d_in[29],
              *rb1 = (const float*)d_in[30], *rb2 = (const float*)d_in[31];
  const float *sW1 = (const float*)d_in[32], *sW2 = (const float*)d_in[33],
              *sb1 = (const float*)d_in[34], *sb2 = (const float*)d_in[35];
  const float *uW1 = (const float*)d_in[36], *uW2 = (const float*)d_in[37],
              *ub1 = (const float*)d_in[38], *ub2 = (const float*)d_in[39];

  // output regions
  float* out   = (float*)d_out;
  float* resp  = out;
  float* surv  = out + NT;
  float* unc   = out + 2 * (size_t)NT;
  float* h2p   = out + 3 * (size_t)NT;                      // [NP,32]
  float* h2t   = out + 3 * (size_t)NT + (size_t)NP * 32;    // [NT,32]

  // workspace carve-up (floats)
  float* ws = (float*)d_ws;
  size_t off = 0;
  auto alloc = [&](size_t n) { float* p = ws + off; off += n; return p; };
  float* hp    = alloc((size_t)NP * 32);
  float* ht    = alloc((size_t)NT * 32);
  float* hs1pt = alloc((size_t)NP * 128);
  float* hd1pt = alloc((size_t)NT * 128);
  float* hs1tp = alloc((size_t)NT * 128);
  float* hd1tp = alloc((size_t)NP * 128);
  float* h1t   = alloc((size_t)NT * 128);
  float* h1p   = alloc((size_t)NP * 128);
  float* hs2pt = alloc((size_t)NP * 32);
  float* hd2pt = alloc((size_t)NT * 32);
  float* hs2tp = alloc((size_t)NT * 32);
  float* hd2tp = alloc((size_t)NP * 32);
  float* as1pt = alloc((size_t)NP * 4);
  float* ad1pt = alloc((size_t)NT * 4);
  float* as1tp = alloc((size_t)NT * 4);
  float* ad1tp = alloc((size_t)NP * 4);
  float* as2pt = alloc((size_t)NP);
  float* ad2pt = alloc((size_t)NT);
  float* as2tp = alloc((size_t)NT);
  float* ad2tp = alloc((size_t)NP);
  float* mx1pt = alloc((size_t)NT * 4);
  float* dn1pt = alloc((size_t)NT * 4);
  float* mx1tp = alloc((size_t)NP * 4);
  float* dn1tp = alloc((size_t)NP * 4);
  float* mx2pt = alloc((size_t)NT);
  float* dn2pt = alloc((size_t)NT);
  float* mx2tp = alloc((size_t)NP);
  float* dn2tp = alloc((size_t)NP);
  (void)ws_size; (void)n_in; (void)out_size;

  const float NEG_INF = -__builtin_huge_valf();
  auto fill = [&](float* p, float v, size_t n) {
    int g = (int)(((n + 255) / 256) < 4096 ? ((n + 255) / 256) : 4096);
    fill_kernel<<<g, 256, 0, stream>>>(p, v, n);
  };
  auto cdiv = [](long a, long b) { return (int)((a + b - 1) / b); };
  // grid for the B-resident GEMM: waves = ceil(M/16 / MT) * (N/16), 8 waves/block
  constexpr int MT = 4;
  auto gemm_grid = [&](int M, int N) {
    long waves = (long)cdiv(M / 16, MT) * (N / 16);
    return cdiv(waves, 8);
  };

  // --- init accumulators (must be per-call: harness does not re-poison) ---
  fill(h1t, 0.0f, (size_t)NT * 128);
  fill(h1p, 0.0f, (size_t)NP * 128);
  fill(h2t, 0.0f, (size_t)NT * 32);
  fill(h2p, 0.0f, (size_t)NP * 32);
  fill(mx1pt, NEG_INF, (size_t)NT * 4);  fill(dn1pt, 0.0f, (size_t)NT * 4);
  fill(mx1tp, NEG_INF, (size_t)NP * 4);  fill(dn1tp, 0.0f, (size_t)NP * 4);
  fill(mx2pt, NEG_INF, (size_t)NT);      fill(dn2pt, 0.0f, (size_t)NT);
  fill(mx2tp, NEG_INF, (size_t)NP);      fill(dn2tp, 0.0f, (size_t)NP);

  // --- encoders (WMMA f32) ---
  gemm_wmma_kernel<128, 32, true, false, MT><<<gemm_grid(NP, 32), 256, 0, stream>>>(x_p, enc_p_W, enc_p_b, hp, NP);
  gemm_wmma_kernel<64,  32, true, false, MT><<<gemm_grid(NT, 32), 256, 0, stream>>>(x_t, enc_t_W, enc_t_b, ht, NT);

  // --- conv1 projections: [*,32] @ [32,128] ---
  gemm_wmma_kernel<32, 128, false, false, MT><<<gemm_grid(NP, 128), 256, 0, stream>>>(hp, c1pt_Ws, nullptr, hs1pt, NP);
  gemm_wmma_kernel<32, 128, false, false, MT><<<gemm_grid(NT, 128), 256, 0, stream>>>(ht, c1pt_Wd, nullptr, hd1pt, NT);
  gemm_wmma_kernel<32, 128, false, false, MT><<<gemm_grid(NT, 128), 256, 0, stream>>>(ht, c1tp_Ws, nullptr, hs1tp, NT);
  gemm_wmma_kernel<32, 128, false, false, MT><<<gemm_grid(NP, 128), 256, 0, stream>>>(hp, c1tp_Wd, nullptr, hd1tp, NP);

  // --- conv1 attention logits (H=4) ---
  attn_logits_kernel<4><<<cdiv((long)NP * 4, 256), 256, 0, stream>>>(hs1pt, c1pt_as, as1pt, NP);
  attn_logits_kernel<4><<<cdiv((long)NT * 4, 256), 256, 0, stream>>>(hd1pt, c1pt_ad, ad1pt, NT);
  attn_logits_kernel<4><<<cdiv((long)NT * 4, 256), 256, 0, stream>>>(hs1tp, c1tp_as, as1tp, NT);
  attn_logits_kernel<4><<<cdiv((long)NP * 4, 256), 256, 0, stream>>>(hd1tp, c1tp_ad, ad1tp, NP);

  // --- conv1 edge softmax + scatter: pt (dst = treatments) ---
  {
    const int *src = ei_pt, *dst = ei_pt + E;
    edge_max_kernel<4><<<cdiv(E, 256), 256, 0, stream>>>(src, dst, as1pt, ad1pt, mx1pt, E);
    edge_expsum_kernel<4><<<cdiv(E, 256), 256, 0, stream>>>(src, dst, as1pt, ad1pt, mx1pt, dn1pt, E);
    edge_scatter_kernel<4><<<cdiv((long)E * 4, 256), 256, 0, stream>>>(src, dst, as1pt, ad1pt, mx1pt, dn1pt, hs1pt, h1t, E);
  }
  // --- conv1 edge softmax + scatter: tp (dst = patients) ---
  {
    const int *src = ei_tp, *dst = ei_tp + E;
    edge_max_kernel<4><<<cdiv(E, 256), 256, 0, stream>>>(src, dst, as1tp, ad1tp, mx1tp, E);
    edge_expsum_kernel<4><<<cdiv(E, 256), 256, 0, stream>>>(src, dst, as1tp, ad1tp, mx1tp, dn1tp, E);
    edge_scatter_kernel<4><<<cdiv((long)E * 4, 256), 256, 0, stream>>>(src, dst, as1tp, ad1tp, mx1tp, dn1tp, hs1tp, h1p, E);
  }
  // bias + relu
  bias_act_kernel<true, 128><<<cdiv((long)NT * 128, 256), 256, 0, stream>>>(h1t, c1pt_b, (size_t)NT * 128);
  bias_act_kernel<true, 128><<<cdiv((long)NP * 128, 256), 256, 0, stream>>>(h1p, c1tp_b, (size_t)NP * 128);

  // --- conv2 projections: [*,128] @ [128,32] ---
  gemm_wmma_kernel<128, 32, false, false, MT><<<gemm_grid(NP, 32), 256, 0, stream>>>(h1p, c2pt_Ws, nullptr, hs2pt, NP);
  gemm_wmma_kernel<128, 32, false, false, MT><<<gemm_grid(NT, 32), 256, 0, stream>>>(h1t, c2pt_Wd, nullptr, hd2pt, NT);
  gemm_wmma_kernel<128, 32, false, false, MT><<<gemm_grid(NT, 32), 256, 0, stream>>>(h1t, c2tp_Ws, nullptr, hs2tp, NT);
  gemm_wmma_kernel<128, 32, false, false, MT><<<gemm_grid(NP, 32), 256, 0, stream>>>(h1p, c2tp_Wd, nullptr, hd2tp, NP);

  // --- conv2 attention logits (H=1) ---
  attn_logits_kernel<1><<<cdiv(NP, 256), 256, 0, stream>>>(hs2pt, c2pt_as, as2pt, NP);
  attn_logits_kernel<1><<<cdiv(NT, 256), 256, 0, stream>>>(hd2pt, c2pt_ad, ad2pt, NT);
  attn_logits_kernel<1><<<cdiv(NT, 256), 256, 0, stream>>>(hs2tp, c2tp_as, as2tp, NT);
  attn_logits_kernel<1><<<cdiv(NP, 256), 256, 0, stream>>>(hd2tp, c2tp_ad, ad2tp, NP);

  // --- conv2 edge softmax + scatter directly into d_out regions ---
  {
    const int *src = ei_pt, *dst = ei_pt + E;
    edge_max_kernel<1><<<cdiv(E, 256), 256, 0, stream>>>(src, dst, as2pt, ad2pt, mx2pt, E);
    edge_expsum_kernel<1><<<cdiv(E, 256), 256, 0, stream>>>(src, dst, as2pt, ad2pt, mx2pt, dn2pt, E);
    edge_scatter_kernel<1><<<cdiv(E, 256), 256, 0, stream>>>(src, dst, as2pt, ad2pt, mx2pt, dn2pt, hs2pt, h2t, E);
  }
  {
    const int *src = ei_tp, *dst = ei_tp + E;
    edge_max_kernel<1><<<cdiv(E, 256), 256, 0, stream>>>(src, dst, as2tp, ad2tp, mx2tp, E);
    edge_expsum_kernel<1><<<cdiv(E, 256), 256, 0, stream>>>(src, dst, as2tp, ad2tp, mx2tp, dn2tp, E);
    edge_scatter_kernel<1><<<cdiv(E, 256), 256, 0, stream>>>(src, dst, as2tp, ad2tp, mx2tp, dn2tp, hs2tp, h2p, E);
  }
  // bias (no activation on conv2 output)
  bias_act_kernel<false, 32><<<cdiv((long)NT * 32, 256), 256, 0, stream>>>(h2t, c2pt_b, (size_t)NT * 32);
  bias_act_kernel<false, 32><<<cdiv((long)NP * 32, 256), 256, 0, stream>>>(h2p, c2tp_b, (size_t)NP * 32);

  // --- prediction heads ---
  heads_kernel<<<cdiv(NT, 256), 256, 0, stream>>>(h2t,
      rW1, rW2, rb1, rb2, sW1, sW2, sb1, sb2, uW1, uW2, ub1, ub2,
      resp, surv, unc, NT);
}